// GraphSageTopK_86045374808915
// MI455X (gfx1250) — compile-verified
//
#include <hip/hip_runtime.h>
#include <math.h>

typedef __attribute__((ext_vector_type(2))) float v2f;
typedef __attribute__((ext_vector_type(8))) float v8f;
typedef unsigned long long ull;

#define NN 100000      // nodes
#define NE 3200000     // edges
#define FIN 128
#define HD 16
#define NPAD 131072    // next pow2 >= NN
#define NTILES (NN / 16)

// ---------------- utility ----------------
__global__ void zero_f32(float* __restrict__ p, long long n) {
  long long i = (long long)blockIdx.x * blockDim.x + threadIdx.x;
  if (i < n) p[i] = 0.0f;
}

// ---------------- conv1 edge scatter: 1 wave per edge, 4 floats/lane ----------------
__global__ void scatter_conv1(const int* __restrict__ ei, const float* __restrict__ x,
                              float* __restrict__ agg, float* __restrict__ cnt, int ne) {
  int t = blockIdx.x * blockDim.x + threadIdx.x;
  int e = t >> 5;
  int lane = t & 31;
  if (e >= ne) return;
  int src = ei[e];
  int dst = ei[ne + e];
  const float4 v = ((const float4*)(x + (size_t)src * FIN))[lane];
  float* d = agg + (size_t)dst * FIN + lane * 4;
  atomicAdd(d + 0, v.x);
  atomicAdd(d + 1, v.y);
  atomicAdd(d + 2, v.z);
  atomicAdd(d + 3, v.w);
  if (lane == 0) atomicAdd(cnt + dst, 1.0f);
}

// ---------------- H=16 edge scatter: 16 threads per edge ----------------
__global__ void scatter_h16(const int* __restrict__ ei, const float* __restrict__ xin,
                            float* __restrict__ agg, float* __restrict__ cnt, int ne) {
  int t = blockIdx.x * blockDim.x + threadIdx.x;
  int e = t >> 4;
  int f = t & 15;
  if (e >= ne) return;
  int src = ei[e];
  int dst = ei[ne + e];
  atomicAdd(&agg[(size_t)dst * HD + f], xin[(size_t)src * HD + f]);
  if (f == 0) atomicAdd(&cnt[dst], 1.0f);
}

// ---------------- fused SAGE GEMM via V_WMMA_F32_16X16X4_F32 ----------------
// out = relu( (agg/max(cnt,1)) @ wl + bl + xin @ wr ), one wave per 16-row tile.
// Weights are staged once per block into LDS (B fragments come from ds_load).
template <int KDIM>
__global__ void sage_gemm(const float* __restrict__ agg, const float* __restrict__ cnt,
                          const float* __restrict__ xin,
                          const float* __restrict__ wl, const float* __restrict__ bl,
                          const float* __restrict__ wr,
                          float* __restrict__ out, int ntiles) {
  __shared__ float lwl[KDIM * 16];
  __shared__ float lwr[KDIM * 16];
  for (int i = threadIdx.x; i < KDIM * 16; i += blockDim.x) {
    lwl[i] = wl[i];
    lwr[i] = wr[i];
  }
  __syncthreads();  // before any wave drops out, so the barrier sees the whole block

  int wave = (blockIdx.x * blockDim.x + threadIdx.x) >> 5;
  if (wave >= ntiles) return;  // wave-uniform: EXEC stays all-ones for WMMA
  int lane = threadIdx.x & 31;
  int m = lane & 15;   // A: row in tile / B,D: column n
  int kh = lane >> 4;  // K-half selector
  int row = wave * 16 + m;

  float icnt = 1.0f / fmaxf(cnt[row], 1.0f);
  const float* arow = agg + (size_t)row * KDIM;
  const float* xrow = xin + (size_t)row * KDIM;

  v8f c = {0.f, 0.f, 0.f, 0.f, 0.f, 0.f, 0.f, 0.f};
  for (int k0 = 0; k0 < KDIM; k0 += 4) {
    int ka = k0 + 2 * kh;
    v2f a0, b0, a1, b1;
    a0.x = arow[ka] * icnt;
    a0.y = arow[ka + 1] * icnt;
    b0.x = lwl[ka * 16 + m];
    b0.y = lwl[(ka + 1) * 16 + m];
    c = __builtin_amdgcn_wmma_f32_16x16x4_f32(false, a0, false, b0, (short)0, c, false, false);
    a1.x = xrow[ka];
    a1.y = xrow[ka + 1];
    b1.x = lwr[ka * 16 + m];
    b1.y = lwr[(ka + 1) * 16 + m];
    c = __builtin_amdgcn_wmma_f32_16x16x4_f32(false, a1, false, b1, (short)0, c, false, false);
  }
  float bias = bl[m];
#pragma unroll
  for (int v = 0; v < 8; v++) {
    int orow = wave * 16 + v + 8 * kh;  // D layout: VGPR v -> M = v + 8*(lane/16)
    out[(size_t)orow * HD + m] = fmaxf(c[v] + bias, 0.0f);
  }
}

// ---------------- TopK pool: score + pack sortable keys ----------------
__global__ void score_pack(const float* __restrict__ h, const float* __restrict__ w,
                           float* __restrict__ score, ull* __restrict__ sbuf, int n, int npad) {
  int i = blockIdx.x * blockDim.x + threadIdx.x;
  if (i >= npad) return;
  if (i < n) {
    float s = 0.f, ww = 0.f;
#pragma unroll
    for (int j = 0; j < HD; j++) {
      s += h[(size_t)i * HD + j] * w[j];
      ww += w[j] * w[j];
    }
    s *= rsqrtf(ww);
    score[i] = s;
    unsigned int b = __float_as_uint(s);
    unsigned int key = (b & 0x80000000u) ? ~b : (b | 0x80000000u);  // ascending transform
    key = ~key;                                                     // descending order
    sbuf[i] = ((ull)key << 32) | (unsigned int)i;
  } else {
    sbuf[i] = 0xFFFFFFFFFFFFFFFFull;  // padding sorts to the end
  }
}

// global compare-exchange stage (used only while j > 256)
__global__ void bitonic_stage(ull* __restrict__ d, int j, int k) {
  int i = blockIdx.x * blockDim.x + threadIdx.x;
  int ixj = i ^ j;
  if (ixj > i) {
    ull a = d[i], b = d[ixj];
    bool up = ((i & k) == 0);
    if ((a > b) == up) {
      d[i] = b;
      d[ixj] = a;
    }
  }
}

// fused LDS stage: runs j = jstart, jstart/2, ..., 1 over 512-key chunks (jstart <= 256)
__global__ void bitonic_fused(ull* __restrict__ d, int k, int jstart) {
  __shared__ ull s[512];
  int t = threadIdx.x;
  int base = blockIdx.x * 512;
  s[t] = d[base + t];
  s[t + 256] = d[base + t + 256];
  __syncthreads();
  for (int j = jstart; j > 0; j >>= 1) {
    int i = ((t & ~(j - 1)) << 1) | (t & (j - 1));
    int gi = base + i;
    bool up = ((gi & k) == 0);
    ull a = s[i], b = s[i | j];
    if ((a > b) == up) {
      s[i] = b;
      s[i | j] = a;
    }
    __syncthreads();
  }
  d[base + t] = s[t];
  d[base + t + 256] = s[t + 256];
}

// hout[r] = hin[perm[r]] * tanh(score[perm[r]]);  inv[perm[r]] = r
__global__ void gate_permute(const ull* __restrict__ sorted, const float* __restrict__ hin,
                             const float* __restrict__ score, float* __restrict__ hout,
                             int* __restrict__ inv, int n) {
  int t = blockIdx.x * blockDim.x + threadIdx.x;
  int r = t >> 4;
  int f = t & 15;
  if (r >= n) return;
  int p = (int)(sorted[r] & 0xFFFFFFFFull);
  if (f == 0) inv[p] = r;
  hout[(size_t)r * HD + f] = hin[(size_t)p * HD + f] * tanhf(score[p]);
}

__global__ void relabel(const int* __restrict__ ein, const int* __restrict__ inv,
                        int* __restrict__ eout, int n2e) {
  int i = blockIdx.x * blockDim.x + threadIdx.x;
  if (i < n2e) eout[i] = inv[ein[i]];
}

// ---------------- MLP head + log_softmax ----------------
__global__ void mlp_head(const float* __restrict__ h, const float* __restrict__ w1,
                         const float* __restrict__ b1, const float* __restrict__ w2,
                         const float* __restrict__ b2, const float* __restrict__ w3,
                         const float* __restrict__ b3, float* __restrict__ out, int n) {
  int i = blockIdx.x * blockDim.x + threadIdx.x;
  if (i >= n) return;
  float v0[16], v1[16], v2[8], v3[2];
#pragma unroll
  for (int j = 0; j < 16; j++) v0[j] = h[(size_t)i * 16 + j];
#pragma unroll
  for (int o = 0; o < 16; o++) {
    float s = b1[o];
#pragma unroll
    for (int j = 0; j < 16; j++) s += v0[j] * w1[j * 16 + o];
    v1[o] = fmaxf(s, 0.0f);
  }
#pragma unroll
  for (int o = 0; o < 8; o++) {
    float s = b2[o];
#pragma unroll
    for (int j = 0; j < 16; j++) s += v1[j] * w2[j * 8 + o];
    v2[o] = fmaxf(s, 0.0f);
  }
#pragma unroll
  for (int o = 0; o < 2; o++) {
    float s = b3[o];
#pragma unroll
    for (int j = 0; j < 8; j++) s += v2[j] * w3[j * 2 + o];
    v3[o] = s;
  }
  float mx = fmaxf(v3[0], v3[1]);
  float lse = logf(expf(v3[0] - mx) + expf(v3[1] - mx));
  out[(size_t)i * 2 + 0] = v3[0] - mx - lse;
  out[(size_t)i * 2 + 1] = v3[1] - mx - lse;
}

// ---------------- launch ----------------
extern "C" void kernel_launch(void* const* d_in, const int* in_sizes, int n_in,
                              void* d_out, int out_size, void* d_ws, size_t ws_size,
                              hipStream_t stream) {
  const float* x = (const float*)d_in[0];
  const int* ei_in = (const int*)d_in[1];
  // d_in[2] = edge_weight (unused, as in the reference)
  const float* w1l = (const float*)d_in[3];
  const float* b1l = (const float*)d_in[4];
  const float* w1r = (const float*)d_in[5];
  const float* p1w = (const float*)d_in[6];
  const float* w2l = (const float*)d_in[7];
  const float* b2l = (const float*)d_in[8];
  const float* w2r = (const float*)d_in[9];
  const float* p2w = (const float*)d_in[10];
  const float* w3l = (const float*)d_in[11];
  const float* b3l = (const float*)d_in[12];
  const float* w3r = (const float*)d_in[13];
  const float* p3w = (const float*)d_in[14];
  const float* l1w = (const float*)d_in[15];
  const float* l1b = (const float*)d_in[16];
  const float* l2w = (const float*)d_in[17];
  const float* l2b = (const float*)d_in[18];
  const float* l3w = (const float*)d_in[19];
  const float* l3b = (const float*)d_in[20];
  float* out = (float*)d_out;

  // workspace carve-up (~124 MB)
  float* agg1 = (float*)d_ws;             // NN*128
  float* cnt1 = agg1 + (size_t)NN * FIN;  // NN   (contiguous with agg1 for one-shot zero)
  float* hA = cnt1 + NN;                  // NN*16
  float* hB = hA + (size_t)NN * HD;       // NN*16
  float* score = hB + (size_t)NN * HD;    // NN
  ull* sbuf = (ull*)(score + NN);         // NPAD keys (offset is a multiple of 8 bytes)
  int* inv = (int*)(sbuf + NPAD);         // NN
  int* eiA = inv + NN;                    // 2*NE
  int* eiB = eiA + (size_t)2 * NE;        // 2*NE
  float* agg2 = (float*)(eiB + (size_t)2 * NE);  // NN*16
  float* cnt2 = agg2 + (size_t)NN * HD;          // NN (contiguous with agg2)

  const int B = 256;
  auto run_pool = [&](const float* hin, const float* pw, float* hout) {
    score_pack<<<NPAD / B, B, 0, stream>>>(hin, pw, score, sbuf, NN, NPAD);
    for (int k = 2; k <= NPAD; k <<= 1) {
      for (int j = k >> 1; j > 256; j >>= 1)
        bitonic_stage<<<NPAD / B, B, 0, stream>>>(sbuf, j, k);
      int jstart = (k >> 1) < 256 ? (k >> 1) : 256;
      bitonic_fused<<<NPAD / 512, 256, 0, stream>>>(sbuf, k, jstart);
    }
    gate_permute<<<(NN * 16 + B - 1) / B, B, 0, stream>>>(sbuf, hin, score, hout, inv, NN);
  };

  // ---- conv1 (F=128 -> H=16) ----
  {
    long long nz = (long long)NN * (FIN + 1);
    zero_f32<<<(unsigned)((nz + B - 1) / B), B, 0, stream>>>(agg1, nz);
    scatter_conv1<<<(NE * 32) / B, B, 0, stream>>>(ei_in, x, agg1, cnt1, NE);
    sage_gemm<FIN><<<(NTILES * 32 + B - 1) / B, B, 0, stream>>>(agg1, cnt1, x, w1l, b1l, w1r, hA,
                                                                NTILES);
  }
  // ---- pool1 + relabel ----
  run_pool(hA, p1w, hB);
  relabel<<<(2 * NE + B - 1) / B, B, 0, stream>>>(ei_in, inv, eiA, 2 * NE);

  // ---- conv2 (16 -> 16) ----
  {
    long long nz = (long long)NN * (HD + 1);
    zero_f32<<<(unsigned)((nz + B - 1) / B), B, 0, stream>>>(agg2, nz);
    scatter_h16<<<(NE * 16) / B, B, 0, stream>>>(eiA, hB, agg2, cnt2, NE);
    sage_gemm<HD><<<(NTILES * 32 + B - 1) / B, B, 0, stream>>>(agg2, cnt2, hB, w2l, b2l, w2r, hA,
                                                               NTILES);
  }
  // ---- pool2 + relabel ----
  run_pool(hA, p2w, hB);
  relabel<<<(2 * NE + B - 1) / B, B, 0, stream>>>(eiA, inv, eiB, 2 * NE);

  // ---- conv3 (16 -> 16) ----
  {
    long long nz = (long long)NN * (HD + 1);
    zero_f32<<<(unsigned)((nz + B - 1) / B), B, 0, stream>>>(agg2, nz);
    scatter_h16<<<(NE * 16) / B, B, 0, stream>>>(eiB, hB, agg2, cnt2, NE);
    sage_gemm<HD><<<(NTILES * 32 + B - 1) / B, B, 0, stream>>>(agg2, cnt2, hB, w3l, b3l, w3r, hA,
                                                               NTILES);
  }
  // ---- pool3 (relabeled edges unused afterwards) ----
  run_pool(hA, p3w, hB);

  // ---- MLP head + log_softmax ----
  mlp_head<<<(NN + B - 1) / B, B, 0, stream>>>(hB, l1w, l1b, l2w, l2b, l3w, l3b, out, NN);
}